// LinkPredictor_83356725281166
// MI455X (gfx1250) — compile-verified
//
#include <hip/hip_runtime.h>
#include <hip/hip_bf16.h>

typedef __attribute__((ext_vector_type(2))) float v2f;
typedef __attribute__((ext_vector_type(8))) float v8f;

#define HIDC 128

// ---------------------------------------------------------------- degree
__global__ void deg_kernel(const long long* __restrict__ dst, float* __restrict__ deg, int E) {
    int e = blockIdx.x * blockDim.x + threadIdx.x;
    if (e < E) atomicAdd(&deg[(int)dst[e]], 1.0f);
}

// ------------------------------------------------- edge scatter (agg += x[src])
__global__ void scatter_kernel(const long long* __restrict__ src,
                               const long long* __restrict__ dst,
                               const float* __restrict__ x,
                               float* __restrict__ agg, int E) {
    int gid  = blockIdx.x * blockDim.x + threadIdx.x;
    int e    = gid >> 5;
    int lane = gid & 31;
    if (e < E) {
        int s = (int)src[e];
        int d = (int)dst[e];
        float4 v = *(const float4*)(x + (size_t)s * HIDC + lane * 4);
        float* o = agg + (size_t)d * HIDC + lane * 4;
        atomicAdd(o + 0, v.x);
        atomicAdd(o + 1, v.y);
        atomicAdd(o + 2, v.z);
        atomicAdd(o + 3, v.w);
    }
}

// Stage a 128x128 row-major weight matrix into LDS, pre-swizzled so that the
// WMMA B-fragment for tile (kt,nt) and lane l is one contiguous 8-byte pair:
//   lds[ ((kt*8+nt)*32 + l)*2 + {0,1} ] = W[(kt*4 + (l>>4)*2 + {0,1})*128 + nt*16 + (l&15)]
// Inner loop then needs a single ds_load_b64 per fragment (aligned VGPR pair).
__device__ __forceinline__ void stage_swizzled(const float* __restrict__ W, float* lds) {
    for (int item = threadIdx.x; item < 2048; item += 256) {
        int ep = item >> 5;              // row pair 0..63  (rows 2ep, 2ep+1)
        int cq = item & 31;              // column quad
        int r0 = ep * 2;
        int n0 = cq * 4;
        float4 f0 = *(const float4*)(W + r0 * HIDC + n0);
        float4 f1 = *(const float4*)(W + (r0 + 1) * HIDC + n0);
        int kt    = ep >> 1;
        int half  = ep & 1;
        int nt    = n0 >> 4;
        int lane0 = half * 16 + (n0 & 15);
        int p0 = ((kt * 8 + nt) * 32 + lane0) * 2;
        float4 o0, o1;
        o0.x = f0.x; o0.y = f1.x; o0.z = f0.y; o0.w = f1.y;
        o1.x = f0.z; o1.y = f1.z; o1.z = f0.w; o1.w = f1.w;
        *(float4*)(lds + p0)     = o0;
        *(float4*)(lds + p0 + 4) = o1;
    }
}

// --------- SAGE linear: out = act( (agg/deg) @ w_l + b + xin @ w_r )
// one wave per 16-row stripe, full fp32 WMMA (16x16x4), weights staged in LDS.
__global__ __launch_bounds__(256) void sage_kernel(
    const float* __restrict__ xin, const float* __restrict__ agg,
    const float* __restrict__ deg,
    const float* __restrict__ w_l, const float* __restrict__ w_r,
    const float* __restrict__ bias, float* __restrict__ out,
    int n, int do_relu)
{
    __shared__ float lw[HIDC * HIDC];          // 64 KB, one weight matrix at a time
    const v2f* lwp = (const v2f*)lw;

    int wave = threadIdx.x >> 5;
    int lane = threadIdx.x & 31;
    int m0   = (blockIdx.x * 8 + wave) * 16;
    bool active = (m0 < n);                    // wave-uniform -> EXEC all ones inside

    int row   = m0 + (lane & 15);
    int khalf = (lane >> 4) * 2;               // lanes 0-15: K pair {0,1}; 16-31: {2,3}

    v8f zero = {};
    v8f acc[8];
#pragma unroll
    for (int t = 0; t < 8; ++t) acc[t] = zero;

    float rdeg = 0.0f;
    if (active) rdeg = 1.0f / fmaxf(deg[row], 1.0f);

    // ---- stage w_l (swizzled), accumulate mean-path ----
    stage_swizzled(w_l, lw);
    __syncthreads();

    if (active) {
        const float* arow = agg + (size_t)row * HIDC;
        for (int kt = 0; kt < 32; ++kt) {
            int k0 = kt * 4 + khalf;
            v2f a;
            a.x = arow[k0]     * rdeg;
            a.y = arow[k0 + 1] * rdeg;
            int base = kt * 8 * 32 + lane;
#pragma unroll
            for (int nt = 0; nt < 8; ++nt) {
                v2f b = lwp[base + nt * 32];
                acc[nt] = __builtin_amdgcn_wmma_f32_16x16x4_f32(
                    false, a, false, b, (short)0, acc[nt], false, false);
            }
        }
    }
    __syncthreads();

    // ---- stage w_r (swizzled), accumulate self-path ----
    stage_swizzled(w_r, lw);
    __syncthreads();

    if (active) {
        const float* arow = xin + (size_t)row * HIDC;
        for (int kt = 0; kt < 32; ++kt) {
            int k0 = kt * 4 + khalf;
            v2f a;
            a.x = arow[k0];
            a.y = arow[k0 + 1];
            int base = kt * 8 * 32 + lane;
#pragma unroll
            for (int nt = 0; nt < 8; ++nt) {
                v2f b = lwp[base + nt * 32];
                acc[nt] = __builtin_amdgcn_wmma_f32_16x16x4_f32(
                    false, a, false, b, (short)0, acc[nt], false, false);
            }
        }

        // epilogue: C tile layout — VGPR v: lanes0-15 row m0+v, lanes16-31 row m0+8+v
        int mbase = m0 + (lane >> 4) * 8;
        int colb  = lane & 15;
#pragma unroll
        for (int nt = 0; nt < 8; ++nt) {
            int col = nt * 16 + colb;
            float bv = bias[col];
#pragma unroll
            for (int v = 0; v < 8; ++v) {
                float val = acc[nt][v] + bv;
                if (do_relu) val = fmaxf(val, 0.0f);
                out[(size_t)(mbase + v) * HIDC + col] = val;
            }
        }
    }
}

// --------- link decoder: out[p] = relu([h[s],h[d]] @ wd1 + bd1) @ wd2 + bd2
__global__ __launch_bounds__(256) void decode_kernel(
    const float* __restrict__ h, const long long* __restrict__ pairs,
    const float* __restrict__ wd1, const float* __restrict__ bd1,
    const float* __restrict__ wd2, const float* __restrict__ bd2,
    float* __restrict__ out, int P)
{
    __shared__ float lw[HIDC * HIDC];          // 64 KB, half of wd1 at a time
    const v2f* lwp = (const v2f*)lw;

    int wave = threadIdx.x >> 5;
    int lane = threadIdx.x & 31;
    int p0   = (blockIdx.x * 8 + wave) * 16;
    bool active = (p0 < P);

    int row   = p0 + (lane & 15);
    int khalf = (lane >> 4) * 2;

    long long sidx = 0, didx = 0;
    if (active) {
        sidx = pairs[row];
        didx = pairs[(size_t)P + row];
    }

    v8f zero = {};
    v8f acc[8];
#pragma unroll
    for (int t = 0; t < 8; ++t) acc[t] = zero;

    // ---- stage wd1 rows 0..127 (source endpoint) ----
    stage_swizzled(wd1, lw);
    __syncthreads();

    if (active) {
        const float* arow = h + (size_t)sidx * HIDC;
        for (int kt = 0; kt < 32; ++kt) {
            int k0 = kt * 4 + khalf;
            v2f a;
            a.x = arow[k0];
            a.y = arow[k0 + 1];
            int base = kt * 8 * 32 + lane;
#pragma unroll
            for (int nt = 0; nt < 8; ++nt) {
                v2f b = lwp[base + nt * 32];
                acc[nt] = __builtin_amdgcn_wmma_f32_16x16x4_f32(
                    false, a, false, b, (short)0, acc[nt], false, false);
            }
        }
    }
    __syncthreads();

    // ---- stage wd1 rows 128..255 (dest endpoint) ----
    stage_swizzled(wd1 + HIDC * HIDC, lw);
    __syncthreads();

    if (active) {
        const float* arow = h + (size_t)didx * HIDC;
        for (int kt = 0; kt < 32; ++kt) {
            int k0 = kt * 4 + khalf;
            v2f a;
            a.x = arow[k0];
            a.y = arow[k0 + 1];
            int base = kt * 8 * 32 + lane;
#pragma unroll
            for (int nt = 0; nt < 8; ++nt) {
                v2f b = lwp[base + nt * 32];
                acc[nt] = __builtin_amdgcn_wmma_f32_16x16x4_f32(
                    false, a, false, b, (short)0, acc[nt], false, false);
            }
        }

        // fused epilogue: relu(+bd1) dot wd2, butterfly-reduce over the 16 N-lanes
        int colb = lane & 15;
        float p8[8];
#pragma unroll
        for (int v = 0; v < 8; ++v) p8[v] = 0.0f;
#pragma unroll
        for (int nt = 0; nt < 8; ++nt) {
            int col = nt * 16 + colb;
            float bv = bd1[col];
            float w2 = wd2[col];
#pragma unroll
            for (int v = 0; v < 8; ++v) {
                float t = fmaxf(acc[nt][v] + bv, 0.0f);
                p8[v] += t * w2;
            }
        }
#pragma unroll
        for (int off = 1; off < 16; off <<= 1) {
#pragma unroll
            for (int v = 0; v < 8; ++v)
                p8[v] += __shfl_xor(p8[v], off, 32);
        }
        if ((lane & 15) == 0) {
            int mb = p0 + (lane >> 4) * 8;       // lane0 -> rows p0..p0+7, lane16 -> +8
            float b2 = bd2[0];
#pragma unroll
            for (int v = 0; v < 8; ++v)
                out[mb + v] = p8[v] + b2;
        }
    }
}

extern "C" void kernel_launch(void* const* d_in, const int* in_sizes, int n_in,
                              void* d_out, int out_size, void* d_ws, size_t ws_size,
                              hipStream_t stream) {
    const float*     x    = (const float*)d_in[0];
    const float*     w1l  = (const float*)d_in[1];
    const float*     b1   = (const float*)d_in[2];
    const float*     w1r  = (const float*)d_in[3];
    const float*     w2l  = (const float*)d_in[4];
    const float*     b2   = (const float*)d_in[5];
    const float*     w2r  = (const float*)d_in[6];
    const float*     wd1  = (const float*)d_in[7];
    const float*     bd1  = (const float*)d_in[8];
    const float*     wd2  = (const float*)d_in[9];
    const float*     bd2  = (const float*)d_in[10];
    const long long* eidx = (const long long*)d_in[11];
    const long long* epr  = (const long long*)d_in[12];
    float* out = (float*)d_out;

    int N = in_sizes[0] / HIDC;
    int E = in_sizes[11] / 2;
    int P = in_sizes[12] / 2;

    float* deg = (float*)d_ws;
    float* agg = deg + N;
    float* h1  = agg + (size_t)N * HIDC;
    float* h2  = h1  + (size_t)N * HIDC;

    int degBlocks  = (E + 255) / 256;
    int scatBlocks = (int)(((size_t)E * 32 + 255) / 256);
    int sageBlocks = (((N + 15) / 16) + 7) / 8;
    int decBlocks  = (((P + 15) / 16) + 7) / 8;

    // zero deg + agg (contiguous)
    hipMemsetAsync(deg, 0, sizeof(float) * ((size_t)N + (size_t)N * HIDC), stream);

    // layer 1
    deg_kernel<<<degBlocks, 256, 0, stream>>>(eidx + E, deg, E);
    scatter_kernel<<<scatBlocks, 256, 0, stream>>>(eidx, eidx + E, x, agg, E);
    sage_kernel<<<sageBlocks, 256, 0, stream>>>(x, agg, deg, w1l, w1r, b1, h1, N, 1);

    // layer 2
    hipMemsetAsync(agg, 0, sizeof(float) * (size_t)N * HIDC, stream);
    scatter_kernel<<<scatBlocks, 256, 0, stream>>>(eidx, eidx + E, h1, agg, E);
    sage_kernel<<<sageBlocks, 256, 0, stream>>>(h1, agg, deg, w2l, w2r, b2, h2, N, 0);

    // decoder
    decode_kernel<<<decBlocks, 256, 0, stream>>>(h2, epr, wd1, bd1, wd2, bd2, out, P);
}